// CalculateMean__24893630447945
// MI455X (gfx1250) — compile-verified
//
#include <hip/hip_runtime.h>
#include <hip/hip_bf16.h>
#include <stdint.h>

// Problem constants (fixed by the reference).
#define C_CLASSES 1000
#define A_DIM     512
#define N_ROWS    262144

// Tiling.
#define A_CHUNK   64                      // features per workgroup (64 floats = 64 LDS banks)
#define N_ACH     (A_DIM / A_CHUNK)       // 8 feature chunks
#define N_SEG     64                      // row segments
#define SEG_ROWS  (N_ROWS / N_SEG)        // 4096 rows per workgroup
#define TILE_ROWS 64                      // rows per staged tile
#define TILES     (SEG_ROWS / TILE_ROWS)  // 64 tiles per workgroup
#define BLOCK     256                     // 8 waves (wave32)

// Dynamic LDS layout (floats):
//   [0, 64000)            : per-class accumulator  (1000 x 64)  = 256 KB
//   [64000, 68096)        : labels for this segment (4096 ints)  = 16 KB
//   [68096, 76288)        : double-buffered staging (2 x 64x64)  = 32 KB
// total = 305,152 bytes  (< 320 KB CDNA5 WGP LDS; impossible on CDNA4)
#define ACC_ELEMS  (C_CLASSES * A_CHUNK)        // 64000
#define LBL_OFF    ACC_ELEMS
#define STG_OFF    (ACC_ELEMS + SEG_ROWS)       // 68096 (16B aligned)
#define STG_ELEMS  (TILE_ROWS * A_CHUNK)        // 4096
#define LDS_FLOATS (STG_OFF + 2 * STG_ELEMS)    // 76288

typedef int           v4i  __attribute__((ext_vector_type(4)));
typedef unsigned int  u32x4 __attribute__((ext_vector_type(4)));
typedef int           i32x4 __attribute__((ext_vector_type(4)));
typedef int           i32x8 __attribute__((ext_vector_type(8)));

// ---- CDNA5 TDM path (tensor_load_to_lds), preferred ----
#if defined(__AMDGCN__) && __has_builtin(__builtin_amdgcn_tensor_load_to_lds)
#define HAVE_TDM 1
#else
#define HAVE_TDM 0
#endif

// ---- CDNA5 async global->LDS path (fallback #1) ----
#if defined(__AMDGCN__) && __has_builtin(__builtin_amdgcn_global_load_async_to_lds_b128)
#define HAVE_ASYNC 1
typedef __attribute__((address_space(1))) v4i GV4;  // global int4
typedef __attribute__((address_space(3))) v4i LV4;  // LDS int4
#else
#define HAVE_ASYNC 0
#endif

#if defined(__AMDGCN__) && __has_builtin(__builtin_amdgcn_s_wait_asynccnt)
#define WAIT_ASYNC(n) __builtin_amdgcn_s_wait_asynccnt(n)
#elif defined(__AMDGCN__)
#define WAIT_ASYNC(n) asm volatile("s_wait_asynccnt %0" ::"i"(n) : "memory")
#else
#define WAIT_ASYNC(n)
#endif

#if defined(__AMDGCN__) && __has_builtin(__builtin_amdgcn_s_wait_tensorcnt)
#define WAIT_TENSOR(n) __builtin_amdgcn_s_wait_tensorcnt(n)
#elif defined(__AMDGCN__)
#define WAIT_TENSOR(n) asm volatile("s_wait_tensorcnt %0" ::"i"(n) : "memory")
#else
#define WAIT_TENSOR(n)
#endif

#if HAVE_TDM
// LDS byte address of a generic pointer into dynamic LDS.
__device__ inline unsigned int lds_byte_addr(const void* p) {
  return (unsigned int)(unsigned long long)
      (__attribute__((address_space(3))) const void*)p;
}
#endif

// -------------------------------------------------------------------------
__global__ void zero_kernel(float* __restrict__ out, float* __restrict__ counts) {
  const int i = blockIdx.x * blockDim.x + threadIdx.x;
  if (i < C_CLASSES * A_DIM) out[i] = 0.0f;
  if (i < C_CLASSES) counts[i] = 0.0f;
}

// Per-block LDS histogram of labels, merged with global float atomics.
__global__ __launch_bounds__(BLOCK)
void count_kernel(const int* __restrict__ labels, float* __restrict__ counts) {
  __shared__ float hist[C_CLASSES];
  for (int i = threadIdx.x; i < C_CLASSES; i += BLOCK) hist[i] = 0.0f;
  __syncthreads();
  const int base = blockIdx.x * SEG_ROWS;
  for (int r = threadIdx.x; r < SEG_ROWS; r += BLOCK)
    atomicAdd(&hist[labels[base + r]], 1.0f);
  __syncthreads();
  for (int i = threadIdx.x; i < C_CLASSES; i += BLOCK)
    atomicAdd(&counts[i], hist[i]);
}

// Main scatter-reduce: block (bx, by) owns feature chunk bx and row segment by.
__global__ __launch_bounds__(BLOCK)
void segsum_kernel(const float* __restrict__ features,
                   const int* __restrict__ labels,
                   float* __restrict__ sums) {
  extern __shared__ float lds[];
  float* acc = lds;                    // [1000][64]
  int*   lbl = (int*)(lds + LBL_OFF);  // [4096]
  float* stg = lds + STG_OFF;          // [2][64][64]

  const int tid  = threadIdx.x;
  const int a0   = blockIdx.x * A_CHUNK;
  const int row0 = blockIdx.y * SEG_ROWS;

  for (int i = tid; i < ACC_ELEMS; i += BLOCK) acc[i] = 0.0f;
  for (int i = tid; i < SEG_ROWS; i += BLOCK) lbl[i] = labels[row0 + i];
  __syncthreads();

  const float* gbase = features + (size_t)row0 * A_DIM + a0;
  const int f  = tid & (A_CHUNK - 1);  // lane -> feature => 32 distinct LDS banks/wave
  const int r0 = tid >> 6;             // 0..3

#if HAVE_TDM
  // --- Tensor Data Mover: one tensor_load_to_lds per 64x64 tile, wave 0 only.
  // D# group1 (constant): wg_mask=0 | data_size=4B(2)<<16 ; dim0=512 ; dim1=4096 ;
  // tile_dim0=64 ; tile_dim1=64 ; dim0_stride=512 ; no pad/iterate/gather.
  i32x8 g1;
  g1[0] = 0x00020000;   // [17:16] data_size = 2 (4 bytes)
  g1[1] = 0x02000000;   // tensor_dim0[15:0]=512 in bits[79:64->31:16]
  g1[2] = 0x10000000;   // tensor_dim1[15:0]=4096 in bits[111:96->31:16]
  g1[3] = 0x00400000;   // tile_dim0=64 in bits[127:112->31:16]
  g1[4] = 64;           // tile_dim1=64 rows
  g1[5] = 512;          // tensor_dim0_stride = 512 elements (row pitch)
  g1[6] = 0;
  g1[7] = 0;
  const i32x4 z4 = {0, 0, 0, 0};
#if __clang_major__ >= 23
  const i32x8 z8 = {0, 0, 0, 0, 0, 0, 0, 0};
#endif
  const unsigned int lds_stage0 = lds_byte_addr(stg);

  auto issue_tdm = [&](int t) {
    const unsigned long long ga =
        (unsigned long long)(uintptr_t)(gbase + (size_t)t * TILE_ROWS * A_DIM);
    u32x4 g0;
    g0[0] = 1u;  // count=1 valid descriptor; not restore/gather
    g0[1] = lds_stage0 + (unsigned int)((t & 1) * (STG_ELEMS * 4));
    g0[2] = (unsigned int)ga;                                    // global_addr[31:0]
    g0[3] = ((unsigned int)(ga >> 32) & 0x01FFFFFFu) | 0x80000000u;  // [56:32] | type=2
#if __clang_major__ >= 23
    __builtin_amdgcn_tensor_load_to_lds(g0, g1, z4, z4, z8, 0);
#else
    __builtin_amdgcn_tensor_load_to_lds(g0, g1, z4, z4, 0);
#endif
  };

  if (tid < 32) issue_tdm(0);
  for (int t = 0; t < TILES; ++t) {
    if (tid < 32) {
      if (t + 1 < TILES) {
        issue_tdm(t + 1);  // prefetch next tile into other buffer
        WAIT_TENSOR(1);    // tensor ops retire in order: tile t complete
      } else {
        WAIT_TENSOR(0);
      }
    }
    __syncthreads();       // staged tile visible to all waves
    const float* l  = stg + (t & 1) * STG_ELEMS;
    const int*   lt = lbl + t * TILE_ROWS;
#pragma unroll
    for (int r = r0; r < TILE_ROWS; r += 4)
      atomicAdd(&acc[lt[r] * A_CHUNK + f], l[r * A_CHUNK + f]);  // ds_add_f32
    __syncthreads();       // buffer free before TDM overwrites it
  }

#elif HAVE_ASYNC
  // --- Async global->LDS staging (8 waves x 4 b128 issues per tile).
  const int s_row = tid >> 4;          // 0..15
  const int s_col = (tid & 15) * 4;    // float index, 16B aligned

  auto issue_tile = [&](int t) {
    const float* g = gbase + (size_t)t * TILE_ROWS * A_DIM;
    float* l = stg + (t & 1) * STG_ELEMS;
#pragma unroll
    for (int k = 0; k < 4; ++k) {
      const int r = s_row + k * 16;
      __builtin_amdgcn_global_load_async_to_lds_b128(
          (GV4*)(g + (size_t)r * A_DIM + s_col),
          (LV4*)(l + r * A_CHUNK + s_col), 0, 0);
    }
  };

  issue_tile(0);
  for (int t = 0; t < TILES; ++t) {
    if (t + 1 < TILES) {
      issue_tile(t + 1);
      WAIT_ASYNC(4);
    } else {
      WAIT_ASYNC(0);
    }
    __syncthreads();
    const float* l  = stg + (t & 1) * STG_ELEMS;
    const int*   lt = lbl + t * TILE_ROWS;
#pragma unroll
    for (int r = r0; r < TILE_ROWS; r += 4)
      atomicAdd(&acc[lt[r] * A_CHUNK + f], l[r * A_CHUNK + f]);
    __syncthreads();
  }
#else
  // --- Fallback: direct streaming (non-temporal) load -> LDS atomic.
  for (int r = r0; r < SEG_ROWS; r += 4) {
    const float v = __builtin_nontemporal_load(gbase + (size_t)r * A_DIM + f);
    atomicAdd(&acc[lbl[r] * A_CHUNK + f], v);
  }
  __syncthreads();
#endif

  // Flush 256 KB accumulator with coalesced global float atomics (L2-resident).
  for (int i = tid; i < ACC_ELEMS; i += BLOCK) {
    atomicAdd(&sums[(size_t)(i >> 6) * A_DIM + a0 + (i & (A_CHUNK - 1))], acc[i]);
  }
}

__global__ void finalize_kernel(float* __restrict__ out,
                                const float* __restrict__ counts) {
  const int i = blockIdx.x * blockDim.x + threadIdx.x;
  if (i < C_CLASSES * A_DIM)
    out[i] = out[i] / fmaxf(counts[i >> 9], 1.0f);  // A_DIM == 512 == 1<<9
}

// -------------------------------------------------------------------------
extern "C" void kernel_launch(void* const* d_in, const int* in_sizes, int n_in,
                              void* d_out, int out_size, void* d_ws, size_t ws_size,
                              hipStream_t stream) {
  (void)in_sizes; (void)n_in; (void)out_size; (void)ws_size;
  const float* features = (const float*)d_in[0];
  const int*   labels   = (const int*)d_in[1];
  float* out    = (float*)d_out;
  float* counts = (float*)d_ws;  // 1000 floats of scratch

  const int total = C_CLASSES * A_DIM;
  zero_kernel<<<(total + BLOCK - 1) / BLOCK, BLOCK, 0, stream>>>(out, counts);
  count_kernel<<<N_SEG, BLOCK, 0, stream>>>(labels, counts);
  segsum_kernel<<<dim3(N_ACH, N_SEG), BLOCK, LDS_FLOATS * sizeof(float), stream>>>(
      features, labels, out);
  finalize_kernel<<<(total + BLOCK - 1) / BLOCK, BLOCK, 0, stream>>>(out, counts);
}